// FRCNN_57990648431223
// MI455X (gfx1250) — compile-verified
//
#include <hip/hip_runtime.h>
#include <stdint.h>

#ifndef __has_builtin
#define __has_builtin(x) 0
#endif

#define B_IMG   8
#define N_PROP  1000
#define N_CLS   80
#define N_LOG   81
#define D_FEAT  2048
#define KC      1024
#define N_T     10
#define NUM_OBJ 6
#define SCORE_TH 1.0e-4f
#define CLS_OFF  4096.0f

// ---------------- workspace layout (uint32 units) ----------------
#define OFF_KEYS   0ull                                  // B * 80000
#define OFF_PROP   (OFF_KEYS + 8ull * 80000ull)          // B * 1024
#define OFF_X1     (OFF_PROP + 8ull * 1024ull)
#define OFF_Y1     (OFF_X1   + 8ull * 1024ull)
#define OFF_X2     (OFF_Y1   + 8ull * 1024ull)
#define OFF_Y2     (OFF_X2   + 8ull * 1024ull)
#define OFF_AREA   (OFF_Y2   + 8ull * 1024ull)
#define OFF_VALID  (OFF_AREA + 8ull * 1024ull)           // B * 32
#define OFF_MASKS  (OFF_VALID + 8ull * 32ull)            // B * 10 * 32 * 1024

// ---------------- CDNA5 async-to-LDS availability ----------------
#if defined(__gfx1250__) && __has_builtin(__builtin_amdgcn_global_load_async_to_lds_b128)
#define HAVE_ASYNC 1
#else
#define HAVE_ASYNC 0
#endif

#if HAVE_ASYNC
typedef int v4i_t __attribute__((ext_vector_type(4)));
__device__ __forceinline__ void async_b128(const uint32_t* gsrc, uint32_t* ldst) {
  auto g = (__attribute__((address_space(1))) v4i_t*)(gsrc);  // global (AS1) int4*
  auto l = (__attribute__((address_space(3))) v4i_t*)(ldst);  // LDS    (AS3) int4*
  __builtin_amdgcn_global_load_async_to_lds_b128(g, l, 0, 0);
}
#if __has_builtin(__builtin_amdgcn_s_wait_asynccnt)
// builtin + explicit compiler fence: staged LDS reads must not hoist above the wait
#define WAIT_ASYNC(n) do { __builtin_amdgcn_s_wait_asynccnt(n); \
                           asm volatile("" ::: "memory"); } while (0)
#else
#define WAIT_ASYNC(n) asm volatile("s_wait_asynccnt " #n ::: "memory")
#endif
#endif

// =====================================================================
// Kernel A: softmax scores -> keys, radix top-1024 select, bitonic sort,
//           candidate boxes (+class offset, clip), validity bitmask.
// One 1024-thread block per image.
// =====================================================================
__global__ __launch_bounds__(1024) void ka_score_select(
    const float* __restrict__ logits, const float* __restrict__ boxes,
    const int* __restrict__ hw, uint32_t* __restrict__ ws)
{
  const int b   = blockIdx.x;
  const int tid = threadIdx.x;
  uint32_t* keys = ws + OFF_KEYS + (size_t)b * (N_PROP * N_CLS);

  // ---- Phase 1: softmax over 81 logits, keys = bits(score) (0 if <= thresh)
  const float* lg = logits + (size_t)b * N_PROP * N_LOG;
  for (int p = tid; p < N_PROP; p += 1024) {
    const float* row = lg + (size_t)p * N_LOG;
    float m = -1e30f;
#pragma unroll 1
    for (int c = 0; c < N_LOG; ++c) m = fmaxf(m, row[c]);
    float s = 0.f;
#pragma unroll 1
    for (int c = 0; c < N_LOG; ++c) s += expf(row[c] - m);
    float inv = 1.0f / s;
#pragma unroll 1
    for (int c = 0; c < N_CLS; ++c) {
      float sc = expf(row[c] - m) * inv;
      keys[p * N_CLS + c] = (sc > SCORE_TH) ? __float_as_uint(sc) : 0u;
    }
  }
  __threadfence();
  __syncthreads();

  // ---- Phase 2: 8-bit radix select of the 1024th-largest nonzero key
  __shared__ uint32_t hist[256];
  __shared__ uint32_t sh_prefix, sh_r, sh_done;
  if (tid == 0) { sh_prefix = 0u; sh_r = KC; sh_done = 0u; }
  __syncthreads();
  for (int shift = 24; shift >= 0; shift -= 8) {
    if (tid < 256) hist[tid] = 0u;
    __syncthreads();
    uint32_t prefix = sh_prefix;
    uint32_t done   = sh_done;
    uint32_t maskAbove = (shift == 24) ? 0u : (0xFFFFFFFFu << (shift + 8));
    if (!done) {
      for (int k = tid; k < N_PROP * N_CLS; k += 1024) {
        uint32_t key = keys[k];
        if (key != 0u && (key & maskAbove) == prefix)
          atomicAdd(&hist[(key >> shift) & 255u], 1u);
      }
    }
    __syncthreads();
    if (tid == 0 && !done) {
      uint32_t r = sh_r;
      if (shift == 24) {
        uint32_t total = 0;
        for (int d = 0; d < 256; ++d) total += hist[d];
        if (total < KC) { sh_prefix = 1u; sh_r = total; sh_done = 1u; }
      }
      if (!sh_done) {
        uint32_t cum = 0; int dSel = 0;
        for (int d = 255; d >= 0; --d) { cum += hist[d]; if (cum >= r) { dSel = d; break; } }
        sh_prefix = prefix | ((uint32_t)dSel << shift);
        sh_r = r - (cum - hist[dSel]);
      }
    }
    __syncthreads();
  }
  const uint32_t P = sh_prefix;   // pivot key; keys >= P qualify (P >= 1)

  // ---- Phase 3: compact qualifying keys into LDS (cap 2048)
  __shared__ uint64_t pairs[2048];
  __shared__ uint32_t sh_cnt;
  if (tid == 0) sh_cnt = 0;
  __syncthreads();
  for (int k = tid; k < N_PROP * N_CLS; k += 1024) {
    uint32_t key = keys[k];
    if (key >= P) {
      uint32_t pos = atomicAdd(&sh_cnt, 1u);
      if (pos < 2048)
        pairs[pos] = ((uint64_t)key << 32) | (uint32_t)(~(uint32_t)k); // key desc, idx asc
    }
  }
  __syncthreads();
  uint32_t cnt = sh_cnt; if (cnt > 2048) cnt = 2048;
  for (int k = tid; k < 2048; k += 1024)
    if (k >= (int)cnt) pairs[k] = 0ull;
  __syncthreads();

  // ---- Phase 4: bitonic sort 2048 descending (composite key)
  for (int k2 = 2; k2 <= 2048; k2 <<= 1) {
    for (int j = k2 >> 1; j > 0; j >>= 1) {
      int i1 = ((tid & ~(j - 1)) << 1) | (tid & (j - 1));
      int i2 = i1 | j;
      uint64_t a = pairs[i1], c = pairs[i2];
      bool desc = ((i1 & k2) == 0);
      if ((a < c) == desc) { pairs[i1] = c; pairs[i2] = a; }
      __syncthreads();
    }
  }

  // ---- Phase 5: emit 1024 candidates (prop, clipped+offset box, area, valid)
  {
    uint64_t pr = pairs[tid];
    uint32_t key = (uint32_t)(pr >> 32);
    uint32_t idx = ~(uint32_t)(pr & 0xFFFFFFFFu);
    bool valid = (key != 0u);
    int prop = 0;
    float x1, y1, x2, y2;
    if (valid) {
      int cls = (int)(idx % N_CLS);
      prop    = (int)(idx / N_CLS);
      const float* bx = boxes + (((size_t)b * N_PROP + prop) * N_CLS + cls) * 4;
      float h = (float)hw[b * 2 + 0], w = (float)hw[b * 2 + 1];
      float off = (float)cls * CLS_OFF;
      x1 = fminf(fmaxf(bx[0], 0.f), w) + off;
      y1 = fminf(fmaxf(bx[1], 0.f), h) + off;
      x2 = fminf(fmaxf(bx[2], 0.f), w) + off;
      y2 = fminf(fmaxf(bx[3], 0.f), h) + off;
    } else {
      x1 = y1 = x2 = y2 = -1.0e9f;   // degenerate far point: IoU == 0 vs anything
    }
    size_t o = (size_t)b * KC + tid;
    float* wf = (float*)ws;
    ws[OFF_PROP + o] = (uint32_t)prop;
    wf[OFF_X1 + o] = x1; wf[OFF_Y1 + o] = y1;
    wf[OFF_X2 + o] = x2; wf[OFF_Y2 + o] = y2;
    wf[OFF_AREA + o] = (x2 - x1) * (y2 - y1);
    uint32_t bal = __builtin_amdgcn_ballot_w32(valid);
    if ((tid & 31) == 0) ws[OFF_VALID + (size_t)b * 32 + (tid >> 5)] = bal;
  }
}

// =====================================================================
// Kernel B: suppression bitmasks for all 10 thresholds in one IoU pass.
// grid (32 row-blocks, 8 images), 1024 threads; wave w owns row rb*32+w.
// Masks stored transposed: [b][t][word(j/32)][row i] for streaming NMS.
// =====================================================================
__global__ __launch_bounds__(1024) void kb_masks(uint32_t* __restrict__ ws)
{
  const int b  = blockIdx.y;
  const int rb = blockIdx.x;
  const int tid = threadIdx.x;
  __shared__ float sx1[KC], sy1[KC], sx2[KC], sy2[KC], sar[KC];
  const float* wf = (const float*)ws;
  size_t cb = (size_t)b * KC;
  sx1[tid] = wf[OFF_X1 + cb + tid];
  sy1[tid] = wf[OFF_Y1 + cb + tid];
  sx2[tid] = wf[OFF_X2 + cb + tid];
  sy2[tid] = wf[OFF_Y2 + cb + tid];
  sar[tid] = wf[OFF_AREA + cb + tid];
  __syncthreads();

  const int wave = tid >> 5, lane = tid & 31;
  const int i = rb * 32 + wave;
  const float rx1 = sx1[i], ry1 = sy1[i], rx2 = sx2[i], ry2 = sy2[i], rar = sar[i];
  uint32_t* masks = ws + OFF_MASKS + (size_t)b * (N_T * 32 * 1024);
  const float TH[N_T] = {0.001f, 0.101f, 0.201f, 0.301f, 0.401f,
                         0.501f, 0.601f, 0.701f, 0.801f, 0.901f};
#pragma unroll 1
  for (int c = 0; c < 32; ++c) {
    int j = c * 32 + lane;
    float iw = fminf(rx2, sx2[j]) - fmaxf(rx1, sx1[j]); iw = fmaxf(iw, 0.f);
    float ih = fminf(ry2, sy2[j]) - fmaxf(ry1, sy1[j]); ih = fmaxf(ih, 0.f);
    float inter = iw * ih;
    float un = rar + sar[j] - inter;
    bool base = (j > i) && (un > 0.f);
    uint32_t wv[N_T];
#pragma unroll
    for (int t = 0; t < N_T; ++t)
      wv[t] = __builtin_amdgcn_ballot_w32(base && (inter > TH[t] * un));  // iou > t
    if (lane == 0) {
#pragma unroll
      for (int t = 0; t < N_T; ++t)
        masks[((size_t)t * 32 + c) * 1024 + i] = wv[t];
    }
  }
}

// =====================================================================
// Kernel C: serial greedy NMS (10 threshold-waves / image), threshold
// selection, top-6, feature gather-transpose. One block per image.
// gfx1250: suppression stream staged via global_load_async_to_lds_b128
// (triple buffer, ASYNCcnt-pipelined); fallback: register ring prefetch.
// =====================================================================
#define PROC(i, w) {                                                        \
    uint32_t kw_ = (uint32_t)__builtin_amdgcn_readlane((int)keep, (i) >> 5);\
    if ((kw_ >> ((i) & 31)) & 1u) keep &= ~(w); }

#define CH 16   // rows staged per chunk (async path)

__global__ __launch_bounds__(1024) void kc_nms_out(
    const float* __restrict__ feats, uint32_t* __restrict__ ws,
    float* __restrict__ out)
{
  const int b = blockIdx.x;
  const int tid = threadIdx.x;
  const int wave = tid >> 5, lane = tid & 31;
  __shared__ uint32_t keepw[N_T * 32];
  __shared__ int counts[N_T];
  __shared__ int sel_prop[NUM_OBJ];
#if HAVE_ASYNC
  __shared__ uint32_t stage[N_T * 3 * 32 * CH];   // [t][buf%3][lane][row-in-chunk]
#endif

  if (wave < N_T) {
    const int t = wave;
    uint32_t keep = ws[OFF_VALID + (size_t)b * 32 + lane];
    const uint32_t* M =
        ws + OFF_MASKS + ((size_t)b * N_T + t) * (32 * 1024) + (size_t)lane * 1024;

#if HAVE_ASYNC
    auto issue_chunk = [&](int c) {
      if (c >= 1024 / CH) return;
      const uint32_t* gs = M + c * CH;
      uint32_t* ls = &stage[((size_t)(t * 3 + (c % 3)) * 32 + lane) * CH];
#pragma unroll
      for (int q = 0; q < CH / 4; ++q) async_b128(gs + q * 4, ls + q * 4);
    };
    issue_chunk(0);
    issue_chunk(1);
#pragma unroll 1
    for (int c = 0; c < 1024 / CH; ++c) {
      issue_chunk(c + 2);                 // writes buf (c+2)%3 (consumed last iter)
      if (c < 1024 / CH - 2)      { WAIT_ASYNC(8); }
      else if (c < 1024 / CH - 1) { WAIT_ASYNC(4); }
      else                        { WAIT_ASYNC(0); }
      const uint32_t* ls = &stage[((size_t)(t * 3 + (c % 3)) * 32 + lane) * CH];
#pragma unroll
      for (int q = 0; q < CH; ++q) { uint32_t w = ls[q]; PROC(c * CH + q, w); }
    }
#else
    // register-ring prefetch: 8 x b128 in flight (32 rows ahead of the serial scan)
    const uint4* Mv = (const uint4*)M;
    uint4 buf[8];
#pragma unroll
    for (int q = 0; q < 8; ++q) buf[q] = Mv[q];
#pragma unroll 1
    for (int g = 0; g < 32; ++g) {
#pragma unroll
      for (int q = 0; q < 8; ++q) {
        int cc = g * 8 + q;
        uint4 cur = buf[q];
        if (cc + 8 < 256) buf[q] = Mv[cc + 8];
        int i0 = cc * 4;
        PROC(i0 + 0, cur.x); PROC(i0 + 1, cur.y);
        PROC(i0 + 2, cur.z); PROC(i0 + 3, cur.w);
      }
    }
#endif
    int myc = __popc(keep);
#pragma unroll
    for (int d = 16; d > 0; d >>= 1) myc += __shfl_down(myc, d, 32);
    keepw[t * 32 + lane] = keep;
    if (lane == 0) counts[t] = myc;
  }
  __syncthreads();

  if (tid == 0) {
    int tsel = N_T - 1;
    for (int t = 0; t < N_T; ++t)
      if (counts[t] >= NUM_OBJ) { tsel = t; break; }
    int ids[NUM_OBJ];
    int nsel = 0;
    for (int i = 0; i < KC && nsel < NUM_OBJ; ++i)
      if ((keepw[tsel * 32 + (i >> 5)] >> (i & 31)) & 1u) ids[nsel++] = i;
    for (int i = 0; i < KC && nsel < NUM_OBJ; ++i)          // -inf filler tie-break
      if (!((keepw[tsel * 32 + (i >> 5)] >> (i & 31)) & 1u)) ids[nsel++] = i;
    for (int k = 0; k < NUM_OBJ; ++k)
      sel_prop[k] = (int)ws[OFF_PROP + (size_t)b * KC + ids[k]];
  }
  __syncthreads();

  // out[b][d][k] = feats[b][prop_k][d]   (coalesced writes)
  const float* fb = feats + (size_t)b * N_PROP * D_FEAT;
  float* ob = out + (size_t)b * D_FEAT * NUM_OBJ;
  for (int e = tid; e < D_FEAT * NUM_OBJ; e += 1024) {
    int d = e / NUM_OBJ;
    int k = e - d * NUM_OBJ;
    ob[e] = fb[(size_t)sel_prop[k] * D_FEAT + d];
  }
}

// =====================================================================
extern "C" void kernel_launch(void* const* d_in, const int* in_sizes, int n_in,
                              void* d_out, int out_size, void* d_ws, size_t ws_size,
                              hipStream_t stream) {
  (void)in_sizes; (void)n_in; (void)out_size; (void)ws_size;
  const float* logits = (const float*)d_in[0];   // (8,1000,81)
  const float* boxes  = (const float*)d_in[1];   // (8,1000,80,4)
  const float* feats  = (const float*)d_in[2];   // (8,1000,2048)
  const int*   hw     = (const int*)d_in[3];     // (8,2)
  uint32_t* ws = (uint32_t*)d_ws;                // ~12.7 MB used
  float* out = (float*)d_out;                    // (8,2048,6)

  ka_score_select<<<B_IMG, 1024, 0, stream>>>(logits, boxes, hw, ws);
  kb_masks<<<dim3(32, B_IMG), 1024, 0, stream>>>(ws);
  kc_nms_out<<<B_IMG, 1024, 0, stream>>>(feats, ws, out);
}